// MANN_LSTMCell_26431228740367
// MI455X (gfx1250) — compile-verified
//
#include <hip/hip_runtime.h>

// MANN LSTM-cell step for MI455X (gfx1250, wave32).
// Bandwidth-bound (~160MB HBM traffic @ 23.3TB/s ~= 7us); all GEMMs run on
// V_WMMA_F32_16X16X4_F32 (true fp32 matrix op) with fused epilogues.
// Streaming reads use global_prefetch_b8; contiguous fragment pairs load as b64.
// Workspace usage ~84.4 MB.

#define MEMN   65536
#define UNITS  256
#define IN_DIM 512
#define BATCH  32

typedef __attribute__((ext_vector_type(2))) float v2f;
typedef __attribute__((ext_vector_type(8))) float v8f;

// D = A(16x4) * B(4x16) + C(16x16), all fp32.
__device__ __forceinline__ v8f wmma_f32(v2f a, v2f b, v8f c) {
  return __builtin_amdgcn_wmma_f32_16x16x4_f32(false, a, false, b, (short)0, c,
                                               false, false);
}

__device__ __forceinline__ v2f ld2(const float* __restrict__ p) {
  return *(const v2f*)p;  // callers guarantee 8B alignment (even float index)
}

__device__ __forceinline__ float sigmoidf_(float x) {
  return 1.0f / (1.0f + expf(-x));
}

// ---------------------------------------------------------------- init ----
__global__ void k_init(float* __restrict__ out,
                       unsigned long long* __restrict__ pmin) {
  const int i = blockIdx.x * 256 + threadIdx.x;
  if (i < BATCH * UNITS) out[i] = 0.0f;
  if (i < BATCH) pmin[i] = ~0ULL;
}

// ------------------------------------------- controller GEMM: z (32x1024) --
// z = [inputs|read] @ kernel + h @ rkernel + bias. One wave per 16-col tile.
// Three straight-line K segments (uniform operand source per segment) so the
// inner loops carry no divergence around the WMMAs.
__global__ void k_controller(const float* __restrict__ inp,
                             const float* __restrict__ rd,
                             const float* __restrict__ h,
                             const float* __restrict__ kw,
                             const float* __restrict__ rkw,
                             const float* __restrict__ bias,
                             float* __restrict__ z) {
  const int lane = threadIdx.x;
  const int n0 = blockIdx.x * 16;      // 64 blocks over 1024 cols
  const int nl = lane & 15, hi = lane >> 4;
  v8f acc0 = {}, acc1 = {};            // batch rows 0-15 / 16-31

  // segment 1: A = inputs (K 0..511), W = kernel rows 0..511
  for (int k = 0; k < IN_DIM; k += 4) {
    const int kk = k + hi * 2;
    const v2f a0 = ld2(&inp[nl * IN_DIM + kk]);
    const v2f a1 = ld2(&inp[(16 + nl) * IN_DIM + kk]);
    v2f b;
    b.x = kw[kk * (4 * UNITS) + n0 + nl];
    b.y = kw[(kk + 1) * (4 * UNITS) + n0 + nl];
    acc0 = wmma_f32(a0, b, acc0);
    acc1 = wmma_f32(a1, b, acc1);
  }
  // segment 2: A = read (K 512..767), W = kernel rows 512..767
  for (int k = 0; k < UNITS; k += 4) {
    const int kk = k + hi * 2;
    const v2f a0 = ld2(&rd[nl * UNITS + kk]);
    const v2f a1 = ld2(&rd[(16 + nl) * UNITS + kk]);
    v2f b;
    b.x = kw[(IN_DIM + kk) * (4 * UNITS) + n0 + nl];
    b.y = kw[(IN_DIM + kk + 1) * (4 * UNITS) + n0 + nl];
    acc0 = wmma_f32(a0, b, acc0);
    acc1 = wmma_f32(a1, b, acc1);
  }
  // segment 3: A = h (K 768..1023), W = recurrent_kernel
  for (int k = 0; k < UNITS; k += 4) {
    const int kk = k + hi * 2;
    const v2f a0 = ld2(&h[nl * UNITS + kk]);
    const v2f a1 = ld2(&h[(16 + nl) * UNITS + kk]);
    v2f b;
    b.x = rkw[kk * (4 * UNITS) + n0 + nl];
    b.y = rkw[(kk + 1) * (4 * UNITS) + n0 + nl];
    acc0 = wmma_f32(a0, b, acc0);
    acc1 = wmma_f32(a1, b, acc1);
  }

  const float bv = bias[n0 + nl];
#pragma unroll
  for (int r = 0; r < 8; ++r) {
    z[(r + hi * 8) * (4 * UNITS) + n0 + nl]      = acc0[r] + bv;
    z[(16 + r + hi * 8) * (4 * UNITS) + n0 + nl] = acc1[r] + bv;
  }
}

// --------------------------------- LSTM gates + key_list + normalized keyT --
__global__ void k_gates(const float* __restrict__ z, const float* __restrict__ cst,
                        float* __restrict__ key, float* __restrict__ nkeyT) {
  const int b = blockIdx.x, u = threadIdx.x;  // 32 blocks x 256 threads
  const float zi = z[b * 1024 + u], zf = z[b * 1024 + 256 + u];
  const float zc = z[b * 1024 + 512 + u], zo = z[b * 1024 + 768 + u];
  const float cn = sigmoidf_(zf) * cst[b * UNITS + u] + sigmoidf_(zi) * tanhf(zc);
  const float kv = sigmoidf_(zo) * tanhf(cn);
  key[b * UNITS + u] = kv;
  __shared__ float red[UNITS];
  red[u] = kv * kv;
  __syncthreads();
  for (int s = 128; s > 0; s >>= 1) {
    if (u < s) red[u] += red[u + s];
    __syncthreads();
  }
  const float rn = rsqrtf(fmaxf(red[0], 1e-12f));
  nkeyT[u * BATCH + b] = kv * rn;  // transposed for coalesced B-fragment loads
}

// ------------------------- write_weights + usage + per-batch packed argmin --
__global__ void k_usage(const float* __restrict__ rw, const float* __restrict__ lu,
                        const float* __restrict__ uw, const float* __restrict__ wg,
                        float* __restrict__ ww,
                        unsigned long long* __restrict__ pmin) {
  __shared__ unsigned long long lmin[BATCH];
  const int t = threadIdx.x;
  if (t < BATCH) lmin[t] = ~0ULL;
  __syncthreads();
  const int i = blockIdx.x * 256 + t;     // 0 .. M*B-1, row-major (M,B)
  const int b = i & (BATCH - 1), m = i >> 5;
  const float g = sigmoidf_(wg[b]);
  const float r = rw[i], l = lu[i];
  const float w = g * r + (1.0f - g) * l;
  ww[i] = w;
  const float usage = 0.95f * uw[i] + r + w;  // strictly positive -> bits monotone
  const unsigned long long p =
      ((unsigned long long)__float_as_uint(usage) << 32) | (unsigned)m;
  atomicMin(&lmin[b], p);
  __syncthreads();
  if (t < BATCH) atomicMin(&pmin[t], lmin[t]);
}

// -------------------------------------------------------- pick global slot --
__global__ void k_slot(const unsigned long long* __restrict__ pmin,
                       int* __restrict__ slot) {
  const int b = threadIdx.x;  // 32 lanes
  unsigned best = (unsigned)(pmin[b] >> 32);
  int bb = b;
  for (int off = 16; off > 0; off >>= 1) {
    const unsigned ob = __shfl_xor(best, off, 32);
    const int oi = __shfl_xor(bb, off, 32);
    if (ob < best || (ob == best && oi < bb)) { best = ob; bb = oi; }
  }
  if (b == 0) *slot = (int)(pmin[bb] & 0xFFFFFFFFu);
}

// ------------- memory_new = ww @ key + scale*memory, fused row-norm (rnorm) --
__global__ void k_memwrite(const float* __restrict__ ww, const float* __restrict__ key,
                           const float* __restrict__ mem, const int* __restrict__ slotp,
                           float* __restrict__ memnew, float* __restrict__ rnorm) {
  const int lane = threadIdx.x & 31, wv = threadIdx.x >> 5;
  const int m0 = (blockIdx.x * 4 + wv) * 16;  // 16-row stripe, full 256 cols
  const int nl = lane & 15, hi = lane >> 4;
  const int slot = *slotp;

  // Prefetch this wave's 16KB tile of `memory` (128 cachelines / 32 lanes).
#pragma unroll
  for (int p = 0; p < 4; ++p)
    __builtin_prefetch(&mem[m0 * UNITS + (p * 32 + lane) * 32], 0, 1);

  v2f a[8];  // A tile 16x32 of write_weights, resident for all 16 N-tiles
#pragma unroll
  for (int s = 0; s < 8; ++s)
    a[s] = ld2(&ww[(m0 + nl) * BATCH + s * 4 + hi * 2]);

  float sq[8] = {0.f, 0.f, 0.f, 0.f, 0.f, 0.f, 0.f, 0.f};
  for (int j = 0; j < 16; ++j) {
    const int c0 = j * 16;
    v8f acc = {};
#pragma unroll
    for (int s = 0; s < 8; ++s) {
      const int kk = s * 4 + hi * 2;
      v2f b;
      b.x = key[kk * UNITS + c0 + nl];
      b.y = key[(kk + 1) * UNITS + c0 + nl];
      acc = wmma_f32(a[s], b, acc);
    }
#pragma unroll
    for (int r = 0; r < 8; ++r) {
      const int m = m0 + r + hi * 8;
      const float sc = (m == slot) ? 0.0f : (float)BATCH;
      const float v = acc[r] + sc * mem[m * UNITS + c0 + nl];
      memnew[m * UNITS + c0 + nl] = v;
      sq[r] += v * v;
    }
  }
#pragma unroll
  for (int r = 0; r < 8; ++r) {  // reduce within each 16-lane half
    float s = sq[r];
    s += __shfl_xor(s, 1, 32); s += __shfl_xor(s, 2, 32);
    s += __shfl_xor(s, 4, 32); s += __shfl_xor(s, 8, 32);
    if (nl == 0) rnorm[m0 + r + hi * 8] = rsqrtf(fmaxf(s, 1e-12f));
  }
}

// ---------------- cos = memnew @ nkey^T (scaled by rnorm) + softmax over B --
__global__ void k_cossoftmax(const float* __restrict__ memnew,
                             const float* __restrict__ nkeyT,
                             const float* __restrict__ rnorm,
                             float* __restrict__ w) {
  const int lane = threadIdx.x & 31, wv = threadIdx.x >> 5;
  const int m0 = (blockIdx.x * 4 + wv) * 16;
  const int nl = lane & 15, hi = lane >> 4;

#pragma unroll
  for (int p = 0; p < 4; ++p)  // prefetch the 16KB memnew tile (mostly L2 hits)
    __builtin_prefetch(&memnew[m0 * UNITS + (p * 32 + lane) * 32], 0, 1);

  v8f acc0 = {}, acc1 = {};  // batch cols 0-15 / 16-31
  for (int k = 0; k < UNITS; k += 4) {
    const int kk = k + hi * 2;
    const v2f a = ld2(&memnew[(m0 + nl) * UNITS + kk]);
    v2f b0, b1;
    b0.x = nkeyT[kk * BATCH + nl];        b0.y = nkeyT[(kk + 1) * BATCH + nl];
    b1.x = nkeyT[kk * BATCH + 16 + nl];   b1.y = nkeyT[(kk + 1) * BATCH + 16 + nl];
    acc0 = wmma_f32(a, b0, acc0);
    acc1 = wmma_f32(a, b1, acc1);
  }
#pragma unroll
  for (int r = 0; r < 8; ++r) {
    const int m = m0 + r + hi * 8;
    const float rn = rnorm[m];
    const float v0 = acc0[r] * rn, v1 = acc1[r] * rn;
    float mx = fmaxf(v0, v1);  // softmax over the 32 batch entries of row m
    mx = fmaxf(mx, __shfl_xor(mx, 1, 32));
    mx = fmaxf(mx, __shfl_xor(mx, 2, 32));
    mx = fmaxf(mx, __shfl_xor(mx, 4, 32));
    mx = fmaxf(mx, __shfl_xor(mx, 8, 32));
    const float e0 = expf(v0 - mx), e1 = expf(v1 - mx);
    float sm = e0 + e1;
    sm += __shfl_xor(sm, 1, 32); sm += __shfl_xor(sm, 2, 32);
    sm += __shfl_xor(sm, 4, 32); sm += __shfl_xor(sm, 8, 32);
    w[m * BATCH + nl] = e0 / sm;
    w[m * BATCH + 16 + nl] = e1 / sm;
  }
}

// --------------------- read_new = w^T @ memnew, split-K + atomic accumulate --
__global__ void k_readnew(const float* __restrict__ w,
                          const float* __restrict__ memnew,
                          float* __restrict__ out) {
  const int lane = threadIdx.x & 31, wv = threadIdx.x >> 5;
  const int gid = blockIdx.x * 4 + wv;  // 1024 waves: 16 u-tiles x 64 k-chunks
  const int c0 = (gid & 15) * 16;
  const int kc = gid >> 4;
  const int nl = lane & 15, hi = lane >> 4;
  v8f acc0 = {}, acc1 = {};  // batch rows 0-15 / 16-31
  const int kend = (kc + 1) * 1024;
  for (int k = kc * 1024; k < kend; k += 4) {
    const int kk = k + hi * 2;
    // prefetch 32 K-rows ahead on both streams
    __builtin_prefetch(&w[(kk + 32) * BATCH + nl], 0, 1);
    __builtin_prefetch(&memnew[(kk + 32) * UNITS + c0 + nl], 0, 1);
    v2f a0, a1, b;
    a0.x = w[kk * BATCH + nl];       a0.y = w[(kk + 1) * BATCH + nl];
    a1.x = w[kk * BATCH + 16 + nl];  a1.y = w[(kk + 1) * BATCH + 16 + nl];
    b.x = memnew[kk * UNITS + c0 + nl];
    b.y = memnew[(kk + 1) * UNITS + c0 + nl];
    acc0 = wmma_f32(a0, b, acc0);
    acc1 = wmma_f32(a1, b, acc1);
  }
#pragma unroll
  for (int r = 0; r < 8; ++r) {
    atomicAdd(&out[(r + hi * 8) * UNITS + c0 + nl], acc0[r]);
    atomicAdd(&out[(16 + r + hi * 8) * UNITS + c0 + nl], acc1[r]);
  }
}

extern "C" void kernel_launch(void* const* d_in, const int* in_sizes, int n_in,
                              void* d_out, int out_size, void* d_ws, size_t ws_size,
                              hipStream_t stream) {
  const float* inputs = (const float*)d_in[0];
  const float* readv  = (const float*)d_in[1];
  const float* rw     = (const float*)d_in[2];
  const float* lu     = (const float*)d_in[3];
  const float* uw     = (const float*)d_in[4];
  const float* memory = (const float*)d_in[5];
  const float* h      = (const float*)d_in[6];
  const float* cst    = (const float*)d_in[7];
  const float* wg     = (const float*)d_in[8];
  const float* kw     = (const float*)d_in[9];
  const float* rkw    = (const float*)d_in[10];
  const float* bias   = (const float*)d_in[11];
  float* out = (float*)d_out;
  float* ws  = (float*)d_ws;

  // Workspace layout (float offsets); total ~84.4 MB.
  float* z      = ws;                 // 32*1024
  float* key    = ws + 32768;         // 32*256
  float* nkeyT  = ws + 40960;         // 256*32 (transposed)
  float* ww     = ws + 49152;         // 65536*32
  float* rnorm  = ws + 2146304;       // 65536
  float* wsm    = ws + 2211840;       // 65536*32 softmax weights
  float* memnew = ws + 4308992;       // 65536*256
  unsigned long long* pmin = (unsigned long long*)(ws + 21086208);  // 32 x u64
  int* slot = (int*)(ws + 21086208 + 64);

  k_init<<<32, 256, 0, stream>>>(out, pmin);
  k_controller<<<64, 32, 0, stream>>>(inputs, readv, h, kw, rkw, bias, z);
  k_gates<<<BATCH, UNITS, 0, stream>>>(z, cst, key, nkeyT);
  k_usage<<<(MEMN * BATCH) / 256, 256, 0, stream>>>(rw, lu, uw, wg, ww, pmin);
  k_slot<<<1, 32, 0, stream>>>(pmin, slot);
  k_memwrite<<<MEMN / 64, 128, 0, stream>>>(ww, key, memory, slot, memnew, rnorm);
  k_cossoftmax<<<MEMN / 64, 128, 0, stream>>>(memnew, nkeyT, rnorm, wsm);
  k_readnew<<<256, 128, 0, stream>>>(wsm, memnew, out);
}